// DirectEncodingModel_56453050139044
// MI455X (gfx1250) — compile-verified
//
#include <hip/hip_runtime.h>

// ---------------------------------------------------------------------------
// DirectEncodingModel on MI455X (gfx1250, wave32, WMMA)
//
// 16 batch rows per 256-thread workgroup. The whole growing activation
// buffer acc (16 x 3584 f32) lives in LDS (row stride padded to 3588 floats
// -> 229,632 B; +19,200 B staged idx tables = 248,832 B < 320 KB WGP LDS;
// the 4-float/row skew makes 16-row column gathers bank-conflict-free).
// Grouped GEMMs run on V_WMMA_F32_16X16X4_F32: M=16 rows, N=16 outputs,
// K in chunks of 4 (24 -> 6 chunks, 48 -> 12 chunks). 8 waves x 8 groups
// cover the 64 groups/layer; one barrier per layer. Bias is folded into the
// WMMA accumulator init; tanh uses native V_TANH_F32 when available.
// HBM traffic: x in (32 MB) + out (4 MB) + L2-resident weights -> ~1.6 us
// floor at 23.3 TB/s.
// ---------------------------------------------------------------------------

typedef float v2f __attribute__((ext_vector_type(2)));
typedef float v8f __attribute__((ext_vector_type(8)));

#define ROW_TILE   16
#define ACC_STRIDE 3588   // 3584 cols + 4-float pad: 16B-aligned rows, bank skew
#define IN_COLS    512
#define NG         64     // groups per hidden layer
#define FH         24     // fan-in per hidden group
#define OG         16     // outputs per group
#define GO_G       4
#define GO_F       48

#define ACC_FLOATS (ROW_TILE * ACC_STRIDE)   // 57,408 floats
#define IDX_H      (NG * FH)                  // 1536 per hidden layer
#define IDX_O      (GO_G * GO_F)              // 192
#define IDX_TOTAL  (3 * IDX_H + IDX_O)        // 4800 ints

__device__ __forceinline__ float fast_tanh(float x) {
#if __has_builtin(__builtin_amdgcn_tanhf)
  return __builtin_amdgcn_tanhf(x);          // native v_tanh_f32
#elif __has_builtin(__builtin_amdgcn_tanh_f32)
  return __builtin_amdgcn_tanh_f32(x);
#else
  x = fminf(fmaxf(x, -9.0f), 9.0f);
  float e = __builtin_amdgcn_exp2f(x * 2.8853900817779268f);
  return (e - 1.0f) * __builtin_amdgcn_rcpf(e + 1.0f);
#endif
}

__global__ void __launch_bounds__(256)
direct_encoding_kernel(const float* __restrict__ x,
                       const float* __restrict__ W1, const float* __restrict__ b1,
                       const float* __restrict__ W2, const float* __restrict__ b2,
                       const float* __restrict__ W3, const float* __restrict__ b3,
                       const float* __restrict__ Wo, const float* __restrict__ bo,
                       const int*   __restrict__ idx1, const int* __restrict__ idx2,
                       const int*   __restrict__ idx3, const int* __restrict__ idxo,
                       float*       __restrict__ out)
{
  extern __shared__ float lds[];                 // acc | idx tables
  int* const idx_lds = reinterpret_cast<int*>(&lds[ACC_FLOATS]);

  const int tid  = threadIdx.x;
  const int wave = tid >> 5;
  const int lane = tid & 31;
  const int half = lane >> 4;      // 0: lanes 0-15, 1: lanes 16-31
  const int n    = lane & 15;      // N for B/C/D frags; M for A-frag loads
  const int row_base = blockIdx.x * ROW_TILE;

  // ---- Stage x tile (16 x 512 f32) into LDS columns [0, 512) --------------
  // 2048 float4s, 8 per thread: global_load_b128 -> ds_store_b128.
  for (int i = tid; i < ROW_TILE * (IN_COLS / 4); i += 256) {
    const int r  = i >> 7;         // 128 float4 per row
    const int c4 = i & 127;
    const float4 v =
        reinterpret_cast<const float4*>(x + (size_t)(row_base + r) * IN_COLS)[c4];
    *reinterpret_cast<float4*>(&lds[r * ACC_STRIDE + c4 * 4]) = v;
  }

  // ---- Stage all idx tables into LDS (kills the global-hop in the
  //      idx -> gather dependency chain) ------------------------------------
  for (int i = tid; i < IDX_TOTAL; i += 256) {
    int v;
    if      (i <     IDX_H) v = idx1[i];
    else if (i < 2 * IDX_H) v = idx2[i - IDX_H];
    else if (i < 3 * IDX_H) v = idx3[i - 2 * IDX_H];
    else                    v = idxo[i - 3 * IDX_H];
    idx_lds[i] = v;
  }
  __syncthreads();

  // ---- Three gather-GEMM-tanh layers --------------------------------------
  const float* Ws[3] = {W1, W2, W3};
  const float* bs[3] = {b1, b2, b3};

  for (int layer = 0; layer < 3; ++layer) {
    const float* W    = Ws[layer];
    const float* bias = bs[layer];
    const int*   idx  = idx_lds + layer * IDX_H;
    const int out_base = IN_COLS + layer * (NG * OG);

    for (int gi = 0; gi < 8; ++gi) {
      const int g = wave * 8 + gi;

      // Fold bias into the accumulator: bias depends only on N (=lane&15),
      // so every C row (VGPR) starts at the same value.
      const float bv = bias[g * OG + n];
      v8f c = {bv, bv, bv, bv, bv, bv, bv, bv};

#pragma unroll
      for (int kc = 0; kc < FH / 4; ++kc) {
        const int k0 = kc * 4 + half * 2;   // this half-wave's K pair
        const int i0 = idx[g * FH + k0];
        const int i1 = idx[g * FH + k0 + 1];
        v2f A, Bf;
        // A 16x4 f32: lane M = n, VGPR0 -> K=2*half, VGPR1 -> K=2*half+1
        A.x = lds[n * ACC_STRIDE + i0];
        A.y = lds[n * ACC_STRIDE + i1];
        // B 4x16 f32: lane N = n, same K striping as A
        Bf.x = W[(g * FH + k0) * OG + n];
        Bf.y = W[(g * FH + k0 + 1) * OG + n];
        c = __builtin_amdgcn_wmma_f32_16x16x4_f32(
                false, A, false, Bf, (short)0, c, false, false);
      }

      const int col = out_base + g * OG + n;
#pragma unroll
      for (int v = 0; v < 8; ++v) {
        const int m = v + half * 8;         // C/D layout: M = v + 8*(lane/16)
        lds[m * ACC_STRIDE + col] = fast_tanh(c[v]);
      }
    }
    __syncthreads();
  }

  // ---- Output layer: 4 groups, F=48, no activation, straight to HBM -------
  if (wave < GO_G) {                        // wave-uniform: EXEC stays all-1s
    const int g = wave;
    const int* idx = idx_lds + 3 * IDX_H;

    const float bv = bo[g * OG + n];
    v8f c = {bv, bv, bv, bv, bv, bv, bv, bv};

#pragma unroll
    for (int kc = 0; kc < GO_F / 4; ++kc) {
      const int k0 = kc * 4 + half * 2;
      const int i0 = idx[g * GO_F + k0];
      const int i1 = idx[g * GO_F + k0 + 1];
      v2f A, Bf;
      A.x  = lds[n * ACC_STRIDE + i0];
      A.y  = lds[n * ACC_STRIDE + i1];
      Bf.x = Wo[(g * GO_F + k0) * OG + n];
      Bf.y = Wo[(g * GO_F + k0 + 1) * OG + n];
      c = __builtin_amdgcn_wmma_f32_16x16x4_f32(
              false, A, false, Bf, (short)0, c, false, false);
    }

#pragma unroll
    for (int v = 0; v < 8; ++v) {
      const int m = v + half * 8;
      out[(size_t)(row_base + m) * (GO_G * OG) + g * OG + n] = c[v];
    }
  }
}

extern "C" void kernel_launch(void* const* d_in, const int* in_sizes, int n_in,
                              void* d_out, int out_size, void* d_ws, size_t ws_size,
                              hipStream_t stream) {
  const float* x    = (const float*)d_in[0];
  const float* W1   = (const float*)d_in[1];
  const float* b1   = (const float*)d_in[2];
  const float* W2   = (const float*)d_in[3];
  const float* b2   = (const float*)d_in[4];
  const float* W3   = (const float*)d_in[5];
  const float* b3   = (const float*)d_in[6];
  const float* Wo   = (const float*)d_in[7];
  const float* bo   = (const float*)d_in[8];
  const int*   idx1 = (const int*)d_in[9];
  const int*   idx2 = (const int*)d_in[10];
  const int*   idx3 = (const int*)d_in[11];
  const int*   idxo = (const int*)d_in[12];
  float* out = (float*)d_out;

  const int B = 16384;
  const size_t lds_bytes =
      (size_t)ACC_FLOATS * sizeof(float) + (size_t)IDX_TOTAL * sizeof(int); // 248,832 B

  // Opt in to >64KB dynamic LDS (WGP supports 320KB). Not a stream op, so
  // it is graph-capture safe and deterministic.
  hipFuncSetAttribute((const void*)direct_encoding_kernel,
                      hipFuncAttributeMaxDynamicSharedMemorySize,
                      (int)lds_bytes);

  dim3 grid(B / ROW_TILE);   // 1024 workgroups
  dim3 block(256);           // 8 wave32s
  direct_encoding_kernel<<<grid, block, lds_bytes, stream>>>(
      x, W1, b1, W2, b2, W3, b3, Wo, bo, idx1, idx2, idx3, idxo, out);
}